// SparseConvExportModule_3796751089673
// MI455X (gfx1250) — compile-verified
//
#include <hip/hip_runtime.h>
#include <stdint.h>

// Problem constants (from reference)
#define ZDIM 41
#define YDIM 1024
#define XDIM 1024
#define CIN  32
#define COUT 64
#define KVOL 27

// LDS weight staging: rows padded 64 -> 72 floats so the two half-wave
// K-rows (banks n..n+15 vs +16 offset) are bank-disjoint.
#define WSTRIDE 72
#define LDS_BYTES (KVOL * CIN * WSTRIDE * 4)   // 27*32*72*4 = 248832 B < 320KB

typedef float v2f __attribute__((ext_vector_type(2)));
typedef float v8f __attribute__((ext_vector_type(8)));

__global__ void scatter_grid_kernel(const int* __restrict__ idx,
                                    int* __restrict__ grid, int n) {
    int i = blockIdx.x * blockDim.x + threadIdx.x;
    if (i < n) {
        int zc = idx[i * 4 + 1];
        int yc = idx[i * 4 + 2];
        int xc = idx[i * 4 + 3];
        grid[(zc * YDIM + yc) * XDIM + xc] = i;
    }
}

__global__ __launch_bounds__(512)
void subm_conv_wmma_kernel(const float* __restrict__ feat,
                           const int* __restrict__ idx,
                           const float* __restrict__ weight,
                           const int* __restrict__ grid,
                           float* __restrict__ out,
                           int n, int numTiles) {
    extern __shared__ float ldsW[];
    const int tid = threadIdx.x;

    // ---- Stage all 27*32*64 weights into LDS via gfx1250 async-to-LDS ----
    {
        const unsigned ldsBase = __builtin_amdgcn_groupstaticsize(); // dyn LDS offset
        const int TOT = KVOL * CIN * (COUT / 4);  // 16B chunks: 27*32*16 = 13824
        for (int c = tid; c < TOT; c += (int)blockDim.x) {
            int row  = c >> 4;        // which [K-row] of W (27*32 rows, 64 floats each)
            int part = c & 15;        // 16B chunk within the row
            unsigned dst = ldsBase + (unsigned)(row * WSTRIDE + part * 4) * 4u;
            unsigned src = (unsigned)(row * COUT + part * 4) * 4u;
            asm volatile("global_load_async_to_lds_b128 %0, %1, %2"
                         :: "v"(dst), "v"(src), "s"(weight) : "memory");
        }
        asm volatile("s_wait_asynccnt 0" ::: "memory");
    }
    __syncthreads();

    // ---- Wave-per-tile sparse gather-GEMM ----
    const int lane  = tid & 31;
    const int m     = lane & 15;   // point-in-tile / N-column
    const int half  = lane >> 4;   // which K/M half this lane serves
    const int waveInBlock  = tid >> 5;
    const int wavesPerBlk  = (int)blockDim.x >> 5;
    const int gwave  = blockIdx.x * wavesPerBlk + waveInBlock;
    const int nWaves = gridDim.x * wavesPerBlk;

    for (int t = gwave; t < numTiles; t += nWaves) {
        const int p = t * 16 + m;
        const bool pvalid = (p < n);
        int zc = 0, yc = 0, xc = 0;
        if (pvalid) {
            zc = idx[p * 4 + 1];
            yc = idx[p * 4 + 2];
            xc = idx[p * 4 + 3];
        }

        v8f c0 = {}, c1 = {}, c2 = {}, c3 = {};

        #pragma unroll 1
        for (int k = 0; k < KVOL; ++k) {
            const int dz = k / 9 - 1;
            const int dy = (k / 3) % 3 - 1;
            const int dx = k % 3 - 1;
            const int nz = zc + dz, ny = yc + dy, nx = xc + dx;
            bool inb = pvalid && (nz >= 0) && (nz < ZDIM) &&
                       (ny >= 0) && (ny < YDIM) && (nx >= 0) && (nx < XDIM);
            int nidx = -1;
            if (inb) nidx = grid[(nz * YDIM + ny) * XDIM + nx];
            const bool valid = inb && (nidx >= 0);

            // Sparse skip: with 0.7% occupancy ~23 of 27 taps are dead per tile.
            if (__builtin_amdgcn_ballot_w32(valid) == 0u) continue;

            const float* frow = feat + (size_t)(valid ? nidx : 0) * CIN;
            const int kbase = k * CIN;

            #pragma unroll
            for (int kk = 0; kk < 8; ++kk) {        // K = 32 in steps of 4
                // A[m][K]: lane(h,m) vgpr j holds K = kk*4 + 2h + j (masked gather)
                v2f a; a.x = 0.0f; a.y = 0.0f;
                if (valid) {
                    const float2 av = *(const float2*)(frow + kk * 4 + half * 2);
                    a.x = av.x; a.y = av.y;
                }
                const int krow = (kbase + kk * 4 + half * 2) * WSTRIDE;
                #pragma unroll
                for (int nt = 0; nt < 4; ++nt) {    // 64 outputs in 4 N-tiles
                    v2f b;
                    b.x = ldsW[krow + nt * 16 + m];
                    b.y = ldsW[krow + WSTRIDE + nt * 16 + m];
                    v8f* cc = (nt == 0) ? &c0 : (nt == 1) ? &c1 : (nt == 2) ? &c2 : &c3;
                    *cc = __builtin_amdgcn_wmma_f32_16x16x4_f32(
                              false, a, false, b, (short)0, *cc, false, false);
                }
            }
        }

        // C/D layout: vgpr r -> row (half*8 + r), lane column m (+16*nt)
        #pragma unroll
        for (int r = 0; r < 8; ++r) {
            const int row = t * 16 + half * 8 + r;
            if (row < n) {
                float* orow = out + (size_t)row * COUT;
                orow[ 0 + m] = c0[r];
                orow[16 + m] = c1[r];
                orow[32 + m] = c2[r];
                orow[48 + m] = c3[r];
            }
        }
    }
}

extern "C" void kernel_launch(void* const* d_in, const int* in_sizes, int n_in,
                              void* d_out, int out_size, void* d_ws, size_t ws_size,
                              hipStream_t stream) {
    const float* feat = (const float*)d_in[0];   // [N, 32] f32
    const int*   idx  = (const int*)d_in[1];     // [N, 4]  i32
    const float* w    = (const float*)d_in[2];   // [27, 32, 64] f32
    float* out = (float*)d_out;                  // [N, 64] f32
    const int n = in_sizes[0] / CIN;

    int* grid = (int*)d_ws;                      // dense hash: 41*1024*1024 i32
    const size_t gridBytes = (size_t)ZDIM * YDIM * XDIM * sizeof(int);
    hipMemsetAsync(d_ws, 0xFF, gridBytes, stream);   // grid[:] = -1

    scatter_grid_kernel<<<(n + 255) / 256, 256, 0, stream>>>(idx, grid, n);

    const int numTiles = (n + 15) / 16;
    subm_conv_wmma_kernel<<<dim3(512), dim3(512), LDS_BYTES, stream>>>(
        feat, idx, w, grid, out, n, numTiles);
}